// NeuromorphicSNNEngine_90709709292014
// MI455X (gfx1250) — compile-verified
//
#include <hip/hip_runtime.h>

typedef float v2f __attribute__((ext_vector_type(2)));
typedef float v8f __attribute__((ext_vector_type(8)));

#define SNN_B  262144
#define SNN_IN 6
#define SNN_H  64
#define SNN_T  5

__global__ void snn_zero(unsigned int* c) { *c = 0u; }

__global__ __launch_bounds__(256) void snn_main(const float* __restrict__ x,
                                                const float* __restrict__ W,
                                                const float* __restrict__ bias,
                                                float* __restrict__ out,
                                                unsigned int* __restrict__ total)
{
    constexpr float INV_TAU = 1.0f / 0.02f;   // 1/TAU_M
    constexpr float V_TH    = -0.05f;
    constexpr float V_RESET = -0.065f;

    const int lane  = threadIdx.x & 31;
    const int wave  = threadIdx.x >> 5;
    const int mtile = wave >> 2;              // 0..1 : which 16-row batch tile
    const int ntile = wave & 3;               // 0..3 : which 16-col hidden tile
    const int mbase = blockIdx.x * 32 + mtile * 16;
    const int nbase = ntile * 16;
    const int half  = lane >> 4;              // K-half select (0 or 1)
    const int l     = lane & 15;              // M row (A) / N col (B,C,D)

    // ---- A fragment: 16x4 f32 tile of x. frag0 = K{2h,2h+1}, frag1 = K{4+2h,...} padded
    const float* xrow = x + (size_t)(mbase + l) * SNN_IN;
    v2f a0, a1;
    a0.x = xrow[2 * half + 0];
    a0.y = xrow[2 * half + 1];
    const float x4 = xrow[4], x5 = xrow[5];
    a1.x = half ? 0.0f : x4;                  // K=6,7 are zero padding
    a1.y = half ? 0.0f : x5;

    // ---- B fragment: 4x16 f32 tile of W (row-major IN x H), same K mapping
    const int col = nbase + l;
    v2f b0, b1;
    b0.x = W[(2 * half + 0) * SNN_H + col];
    b0.y = W[(2 * half + 1) * SNN_H + col];
    const float w4 = W[4 * SNN_H + col], w5 = W[5 * SNN_H + col];
    b1.x = half ? 0.0f : w4;
    b1.y = half ? 0.0f : w5;

    // ---- i_in tile = A*B (K = 6 via two 16x16x4 f32 WMMAs)
    v8f acc = {};
    acc = __builtin_amdgcn_wmma_f32_16x16x4_f32(false, a0, false, b0,
                                                (short)0, acc, false, false);
    acc = __builtin_amdgcn_wmma_f32_16x16x4_f32(false, a1, false, b1,
                                                (short)0, acc, false, false);

    // C/D layout: reg r holds M = r + 8*half, N = col for this lane.
    const float bv = bias[col];
    int cnt = 0;
#pragma unroll
    for (int r = 0; r < 8; ++r) {
        const float i_in = acc[r] + bv;
        const int   row  = mbase + r + 8 * half;
        float* o = out + (size_t)row * (SNN_T * SNN_H) + col;
        float v = 0.0f;
#pragma unroll
        for (int t = 0; t < SNN_T; ++t) {
            v += (i_in - v) * INV_TAU;        // LIF Euler step
            const bool sp = v > V_TH;
            cnt += sp ? 1 : 0;
            // Streamed 335MB output, never re-read: bypass cache (TH=NT).
            __builtin_nontemporal_store(sp ? 1.0f : 0.0f, o + (size_t)t * SNN_H);
            v = sp ? V_RESET : v;
        }
    }

    // ---- deterministic integer spike-count reduction
    __shared__ int red[256];
    red[threadIdx.x] = cnt;
    __syncthreads();
    for (int s = 128; s > 0; s >>= 1) {
        if (threadIdx.x < s) red[threadIdx.x] += red[threadIdx.x + s];
        __syncthreads();
    }
    if (threadIdx.x == 0) atomicAdd(total, (unsigned int)red[0]);
}

__global__ void snn_final(const unsigned int* __restrict__ total,
                          float* __restrict__ out)
{
    out[(size_t)SNN_B * SNN_T * SNN_H] = (float)(*total);
}

extern "C" void kernel_launch(void* const* d_in, const int* in_sizes, int n_in,
                              void* d_out, int out_size, void* d_ws, size_t ws_size,
                              hipStream_t stream)
{
    const float* x = (const float*)d_in[0];   // (B, IN)
    const float* W = (const float*)d_in[1];   // (IN, H)
    const float* b = (const float*)d_in[2];   // (H,)
    float* out = (float*)d_out;               // (B, T, H) spikes + [1] total
    unsigned int* total = (unsigned int*)d_ws;

    snn_zero<<<1, 1, 0, stream>>>(total);
    snn_main<<<SNN_B / 32, 256, 0, stream>>>(x, W, b, out, total);
    snn_final<<<1, 1, 0, stream>>>(total, out);
}